// MagGraphConv_23441931501817
// MI455X (gfx1250) — compile-verified
//
#include <hip/hip_runtime.h>
#include <math.h>

// Problem constants (from reference)
#define NB 64      // batch
#define NV 2000    // nodes
#define ND 16      // embedding dim
#define NCI 64     // in channels
#define NCO 64     // out channels

typedef float v2f __attribute__((ext_vector_type(2)));
typedef float v8f __attribute__((ext_vector_type(8)));

static __device__ __forceinline__ v8f wmma_f32(v2f a, v2f b, v8f c) {
    // V_WMMA_F32_16X16X4_F32 : D = A(16x4) * B(4x16) + C(16x16)
    return __builtin_amdgcn_wmma_f32_16x16x4_f32(
        false, a, false, b, (short)0, c, false, false);
}

// ---------------------------------------------------------------------------
// Kernel 0: s = 1 / ||ne^H ne||_F   (== 1/||lap||_F since lap = ne ne^H)
// ---------------------------------------------------------------------------
__global__ void k_scale(const float* __restrict__ nr, const float* __restrict__ ni,
                        float* __restrict__ s_out) {
    __shared__ float red[256];
    int t = threadIdx.x;
    int d = t >> 4, e = t & 15;
    float gr = 0.f, gi = 0.f;
    for (int v = 0; v < NV; ++v) {
        float ard = nr[v * ND + d], aid = ni[v * ND + d];
        float are = nr[v * ND + e], aie = ni[v * ND + e];
        // conj(ne[:,d]) . ne[:,e]
        gr += ard * are + aid * aie;
        gi += ard * aie - aid * are;
    }
    red[t] = gr * gr + gi * gi;
    __syncthreads();
    for (int off = 128; off > 0; off >>= 1) {
        if (t < off) red[t] += red[t + off];
        __syncthreads();
    }
    if (t == 0) s_out[0] = 1.0f / sqrtf(red[0]);
}

// ---------------------------------------------------------------------------
// Kernel 1: wA = w[:,0]-w[:,2], wB = w[:,1]+2*w[:,2]   (re/im pools)
// ---------------------------------------------------------------------------
__global__ void k_wab(const float* __restrict__ wre, const float* __restrict__ wim,
                      float* __restrict__ waR, float* __restrict__ waI,
                      float* __restrict__ wbR, float* __restrict__ wbI) {
    int idx = blockIdx.x * 256 + threadIdx.x;        // < ND*NCI*NCO = 65536
    int d = idx >> 12;
    int rem = idx & 4095;                            // i*64+o
    const float* wr = wre + (size_t)(d * 3) * 4096;
    const float* wi = wim + (size_t)(d * 3) * 4096;
    float r0 = wr[rem], r1 = wr[4096 + rem], r2 = wr[8192 + rem];
    float i0 = wi[rem], i1 = wi[4096 + rem], i2 = wi[8192 + rem];
    waR[idx] = r0 - r2;
    waI[idx] = i0 - i2;
    wbR[idx] = r1 + 2.0f * r2;
    wbI[idx] = i1 + 2.0f * i2;
}

// ---------------------------------------------------------------------------
// Kernel 2: z[b,d,i] = s * sum_v conj(ne[v,d]) * x[b,v,i]
// One block per batch b. WMMA: M=16 (d), N=64 (4 tiles of 16 over i),
// K=2000 (v), 8 waves = 4 N-tiles x 2 K-halves, LDS reduction of halves.
// ---------------------------------------------------------------------------
__global__ void k_z(const float* __restrict__ nr, const float* __restrict__ ni,
                    const float* __restrict__ x, const float* __restrict__ s_ptr,
                    float* __restrict__ zre, float* __restrict__ zim) {
    __shared__ float red[2][8][256];                 // 16 KB
    int b = blockIdx.x;
    int t = threadIdx.x, lane = t & 31, w = t >> 5;
    int r = lane & 15, h = lane >> 4;
    int n0 = (w & 3) * 16;                           // i tile
    int kh = w >> 2;                                 // K half
    const float* xb = x + (size_t)b * NV * NCI;

    v8f accRe = {0.f, 0.f, 0.f, 0.f, 0.f, 0.f, 0.f, 0.f};
    v8f accIm = {0.f, 0.f, 0.f, 0.f, 0.f, 0.f, 0.f, 0.f};

    int vbase = kh * (NV / 2);
    for (int ks = 0; ks < (NV / 2) / 4; ++ks) {
        int va = vbase + 4 * ks + 2 * h;
        int vb = va + 1;
        v2f aRe, aIm, bX;
        aRe.x = nr[va * ND + r];   aRe.y = nr[vb * ND + r];
        aIm.x = ni[va * ND + r];   aIm.y = ni[vb * ND + r];
        bX.x  = xb[va * NCI + n0 + r];
        bX.y  = xb[vb * NCI + n0 + r];
        accRe = wmma_f32(aRe, bX, accRe);
        accIm = wmma_f32(aIm, bX, accIm);
    }
#pragma unroll
    for (int j = 0; j < 8; ++j) {
        red[0][w][j * 32 + lane] = accRe[j];
        red[1][w][j * 32 + lane] = accIm[j];
    }
    __syncthreads();
    if (kh == 0) {
        float s = s_ptr[0];
#pragma unroll
        for (int j = 0; j < 8; ++j) {
            float zr = (red[0][w][j * 32 + lane] + red[0][w + 4][j * 32 + lane]) * s;
            float zi = -(red[1][w][j * 32 + lane] + red[1][w + 4][j * 32 + lane]) * s;
            int dd = j + 8 * h;                      // C/D row: M = vgpr + 8*(lane/16)
            int ii = n0 + r;
            zre[((size_t)b * ND + dd) * NCI + ii] = zr;
            zim[((size_t)b * ND + dd) * NCI + ii] = zi;
        }
    }
}

// ---------------------------------------------------------------------------
// Kernel 3: per node v:  out[b,v,:] = X*A + Y*B + bias  (complex), interleaved
// LDS layout (floats), row stride 68 (=64+4) for conflict-free WMMA reads:
// ---------------------------------------------------------------------------
#define LSTR 68
#define MSZ  (64 * LSTR)          // 4352 floats per 64x64 matrix
#define OFF_X   0
#define OFF_YR  (1 * MSZ)
#define OFF_YI  (2 * MSZ)
#define OFF_AR  (3 * MSZ)         // A_re transposed: [o][i]
#define OFF_AI  (4 * MSZ)
#define OFF_BR  (5 * MSZ)
#define OFF_BI  (6 * MSZ)
#define OFF_NR  (7 * MSZ)         // ne_v re (16)
#define OFF_NI  (7 * MSZ + 16)    // ne_v im (16)
#define OFF_BRE (7 * MSZ + 32)    // bias re (64)
#define OFF_BIM (7 * MSZ + 96)    // bias im (64)
#define LDS_FLOATS (7 * MSZ + 160)

extern __shared__ float lds[];

__global__ void k_main(const float* __restrict__ x,
                       const float* __restrict__ nr, const float* __restrict__ ni_,
                       const float* __restrict__ waR, const float* __restrict__ waI,
                       const float* __restrict__ wbR, const float* __restrict__ wbI,
                       const float* __restrict__ bpr, const float* __restrict__ bpi,
                       const float* __restrict__ zre, const float* __restrict__ zim,
                       float* __restrict__ out) {
    int v = blockIdx.x;
    int t = threadIdx.x;

    float* LX  = lds + OFF_X;
    float* LYr = lds + OFF_YR;
    float* LYi = lds + OFF_YI;
    float* LAr = lds + OFF_AR;
    float* LAi = lds + OFF_AI;
    float* LBr = lds + OFF_BR;
    float* LBi = lds + OFF_BI;
    float* Lnr = lds + OFF_NR;
    float* Lni = lds + OFF_NI;
    float* Lbr = lds + OFF_BRE;
    float* Lbi = lds + OFF_BIM;

    if (t < ND) {
        Lnr[t] = nr[v * ND + t];
        Lni[t] = ni_[v * ND + t];
    }
    __syncthreads();

    // Stage X[b][i] = x[b,v,i]
#pragma unroll
    for (int it = 0; it < 16; ++it) {
        int e = t + 256 * it;
        int b = e >> 6, i = e & 63;
        LX[b * LSTR + i] = x[((size_t)b * NV + v) * NCI + i];
    }

    // bias_v[o] = sum_d ne[v,d] * b_pool[d,o]
    if (t < NCO) {
        float br = 0.f, bi = 0.f;
#pragma unroll
        for (int d = 0; d < ND; ++d) {
            float nrd = Lnr[d], nid = Lni[d];
            float pr = bpr[d * NCO + t], pi = bpi[d * NCO + t];
            br += nrd * pr - nid * pi;
            bi += nrd * pi + nid * pr;
        }
        Lbr[t] = br;
        Lbi[t] = bi;
    }

    // Y[b][i] = sum_d ne[v,d] * z[b,d,i]   (z already scaled, conj folded)
#pragma unroll 2
    for (int it = 0; it < 16; ++it) {
        int e = t + 256 * it;
        int b = e >> 6, i = e & 63;
        float yr = 0.f, yi = 0.f;
#pragma unroll
        for (int d = 0; d < ND; ++d) {
            float nrd = Lnr[d], nid = Lni[d];
            float zr = zre[((size_t)b * ND + d) * NCI + i];
            float zi = zim[((size_t)b * ND + d) * NCI + i];
            yr += nrd * zr - nid * zi;
            yi += nrd * zi + nid * zr;
        }
        LYr[b * LSTR + i] = yr;
        LYi[b * LSTR + i] = yi;
    }

    // A[i][o] = sum_d ne[v,d]*wA[d,i,o] ; B likewise. Stored transposed [o][i].
#pragma unroll 2
    for (int it = 0; it < 16; ++it) {
        int e = t + 256 * it;                        // e = i*64 + o
        int i = e >> 6, o = e & 63;
        float ar = 0.f, ai = 0.f, br = 0.f, bi = 0.f;
#pragma unroll
        for (int d = 0; d < ND; ++d) {
            float nrd = Lnr[d], nid = Lni[d];
            float war = waR[d * 4096 + e], wai = waI[d * 4096 + e];
            ar += nrd * war - nid * wai;
            ai += nrd * wai + nid * war;
            float wbr = wbR[d * 4096 + e], wbi = wbI[d * 4096 + e];
            br += nrd * wbr - nid * wbi;
            bi += nrd * wbi + nid * wbr;
        }
        LAr[o * LSTR + i] = ar;
        LAi[o * LSTR + i] = ai;
        LBr[o * LSTR + i] = br;
        LBi[o * LSTR + i] = bi;
    }
    __syncthreads();

    // WMMA phase: 16 output tiles (4m x 4n); wave -> 1 m-row, 2 n-tiles.
    int lane = t & 31, w = t >> 5;
    int r = lane & 15, h = lane >> 4;
    int m0  = (w & 3) * 16;
    int n0a = (w >> 2) * 32;
    int n0b = n0a + 16;

    v8f re0 = {0.f,0.f,0.f,0.f,0.f,0.f,0.f,0.f};
    v8f im0 = {0.f,0.f,0.f,0.f,0.f,0.f,0.f,0.f};
    v8f re1 = {0.f,0.f,0.f,0.f,0.f,0.f,0.f,0.f};
    v8f im1 = {0.f,0.f,0.f,0.f,0.f,0.f,0.f,0.f};

#pragma unroll 4
    for (int ks = 0; ks < NCI; ks += 4) {
        int kc = ks + 2 * h;
        v2f aX  = *(const v2f*)&LX [(m0 + r) * LSTR + kc];
        v2f aYr = *(const v2f*)&LYr[(m0 + r) * LSTR + kc];
        v2f aYi = *(const v2f*)&LYi[(m0 + r) * LSTR + kc];

        // tile a
        {
            v2f bAr = *(const v2f*)&LAr[(n0a + r) * LSTR + kc];
            v2f bAi = *(const v2f*)&LAi[(n0a + r) * LSTR + kc];
            v2f bBr = *(const v2f*)&LBr[(n0a + r) * LSTR + kc];
            v2f bBi = *(const v2f*)&LBi[(n0a + r) * LSTR + kc];
            v2f bBin; bBin.x = -bBi.x; bBin.y = -bBi.y;
            re0 = wmma_f32(aX,  bAr,  re0);
            re0 = wmma_f32(aYr, bBr,  re0);
            re0 = wmma_f32(aYi, bBin, re0);
            im0 = wmma_f32(aX,  bAi,  im0);
            im0 = wmma_f32(aYr, bBi,  im0);
            im0 = wmma_f32(aYi, bBr,  im0);
        }
        // tile b
        {
            v2f bAr = *(const v2f*)&LAr[(n0b + r) * LSTR + kc];
            v2f bAi = *(const v2f*)&LAi[(n0b + r) * LSTR + kc];
            v2f bBr = *(const v2f*)&LBr[(n0b + r) * LSTR + kc];
            v2f bBi = *(const v2f*)&LBi[(n0b + r) * LSTR + kc];
            v2f bBin; bBin.x = -bBi.x; bBin.y = -bBi.y;
            re1 = wmma_f32(aX,  bAr,  re1);
            re1 = wmma_f32(aYr, bBr,  re1);
            re1 = wmma_f32(aYi, bBin, re1);
            im1 = wmma_f32(aX,  bAi,  im1);
            im1 = wmma_f32(aYr, bBi,  im1);
            im1 = wmma_f32(aYi, bBr,  im1);
        }
    }

    // Epilogue: add bias, write interleaved (re,im) as float2 -> coalesced
    float brA = Lbr[n0a + r], biA = Lbi[n0a + r];
    float brB = Lbr[n0b + r], biB = Lbi[n0b + r];
#pragma unroll
    for (int j = 0; j < 8; ++j) {
        int b = m0 + j + 8 * h;                       // C/D row mapping
        size_t base = ((size_t)b * NV + v) * (2 * NCO);
        v2f o0; o0.x = re0[j] + brA; o0.y = im0[j] + biA;
        *(v2f*)(out + base + 2 * (n0a + r)) = o0;
        v2f o1; o1.x = re1[j] + brB; o1.y = im1[j] + biB;
        *(v2f*)(out + base + 2 * (n0b + r)) = o1;
    }
}

// ---------------------------------------------------------------------------
extern "C" void kernel_launch(void* const* d_in, const int* in_sizes, int n_in,
                              void* d_out, int out_size, void* d_ws, size_t ws_size,
                              hipStream_t stream) {
    const float* x     = (const float*)d_in[0];
    const float* ne_re = (const float*)d_in[1];
    const float* ne_im = (const float*)d_in[2];
    const float* w_re  = (const float*)d_in[3];
    const float* w_im  = (const float*)d_in[4];
    const float* b_re  = (const float*)d_in[5];
    const float* b_im  = (const float*)d_in[6];
    float* out = (float*)d_out;

    float* wsf = (float*)d_ws;
    float* s_scale = wsf;                       // 1 (padded to 256)
    float* zre = wsf + 256;                     // 64*16*64
    float* zim = zre + NB * ND * NCI;
    float* waR = zim + NB * ND * NCI;           // 16*64*64 each
    float* waI = waR + ND * NCI * NCO;
    float* wbR = waI + ND * NCI * NCO;
    float* wbI = wbR + ND * NCI * NCO;

    k_scale<<<1, 256, 0, stream>>>(ne_re, ne_im, s_scale);
    k_wab<<<(ND * NCI * NCO) / 256, 256, 0, stream>>>(w_re, w_im, waR, waI, wbR, wbI);
    k_z<<<NB, 256, 0, stream>>>(ne_re, ne_im, x, s_scale, zre, zim);
    k_main<<<NV, 256, LDS_FLOATS * sizeof(float), stream>>>(
        x, ne_re, ne_im, waR, waI, wbR, wbI, b_re, b_im, zre, zim, out);
}